// IPM_20555713479419
// MI455X (gfx1250) — compile-verified
//
#include <hip/hip_runtime.h>
#include <hip/hip_bf16.h>
#include <math.h>

typedef __attribute__((ext_vector_type(2))) float v2f;
typedef __attribute__((ext_vector_type(8))) float v8f;

#define BEV    200
#define NPIX   (BEV * BEV)      // 40000
#define IMG_H  128
#define IMG_W  352
#define IMG_C  32
#define NCAM   6
#define NBATCH 2
#define TILE   16               // BEV pixels per block (one WMMA M-dim)

// One block: 256 threads = 8 wave32. Handles TILE=16 BEV pixels of one batch.
// Stage 0: P = K @ RT (6 cams) into LDS.
// Stage 1: waves 0..5 each project 16 pixels through their camera's P with
//          v_wmma_f32_16x16x4_f32 (A = [y x 0 1] rows, B = P^T), store pc0..2.
// Stage 2: each wave processes 2 pixels; lane = channel -> 128B coalesced
//          gathers; bilinear blend exactly as reference; max over cameras.
__global__ __launch_bounds__(256) void ipm_bev_kernel(
    const float* __restrict__ images,   // [2,6,128,352,32]
    const float* __restrict__ Ks,       // [12,4,4]
    const float* __restrict__ RTs,      // [12,4,4]
    float* __restrict__ out)            // [2,200,200,32]
{
    __shared__ float sP [NCAM][4][4];       // P = K @ RT per camera
    __shared__ float sPC[NCAM][TILE][3];    // pc0, pc1, pc2 per (cam, pixel)

    const int bid      = blockIdx.x;
    const int tiles    = NPIX / TILE;       // 2500
    const int b        = bid / tiles;
    const int tile     = bid % tiles;
    const int base_pix = tile * TILE;
    const int t        = threadIdx.x;
    const int wave     = t >> 5;
    const int lane     = t & 31;

    // ---------------- Stage 0: P = K @ RT (96 entries) ----------------
    if (t < NCAM * 16) {
        const int cam = t >> 4;
        const int r   = (t >> 2) & 3;
        const int c   = t & 3;
        const int bn  = b * NCAM + cam;
        const float* Kr = Ks  + (size_t)bn * 16 + r * 4;
        const float* Rt = RTs + (size_t)bn * 16;
        float acc =      Kr[0] * Rt[0 * 4 + c];
        acc = fmaf(Kr[1], Rt[1 * 4 + c], acc);
        acc = fmaf(Kr[2], Rt[2 * 4 + c], acc);
        acc = fmaf(Kr[3], Rt[3 * 4 + c], acc);
        sP[cam][r][c] = acc;
    }
    __syncthreads();

    // ---------------- Stage 1: WMMA projection (waves 0..5) ----------------
    if (wave < NCAM) {                       // wave-uniform branch: EXEC all-1s
        const int cam = wave;

        // A-matrix 16x4 f32 (ISA 7.12.2): lanes 0-15 -> M=lane, {K0,K1} in
        // VGPR0/1; lanes 16-31 -> M=lane-16, {K2,K3}.
        // World homogeneous vector is X = (y, x, 0, 1)  [coords stacking order].
        const int   m  = lane & 15;
        const int   gp = base_pix + m;
        const int   iy = gp / BEV;
        const int   ix = gp % BEV;
        const float xf = (float)(-50.0 + (double)ix * (100.0 / 199.0)); // linspace
        const float yf = (float)(-50.0 + (double)iy * (100.0 / 199.0));
        v2f a;
        a.x = (lane < 16) ? yf : 0.0f;   // K0 = y   | K2 = 0
        a.y = (lane < 16) ? xf : 1.0f;   // K1 = x   | K3 = 1

        // B-matrix 4x16 f32: row K striped across lanes within a VGPR:
        // VGPR0 = rows K0 (lanes 0-15) / K2 (lanes 16-31); VGPR1 = K1 / K3.
        // B[k][n] = (P^T)[k][n] = P[n][k]; columns n>=4 are don't-care (0).
        const int n  = lane & 15;
        const int nn = (n < 4) ? n : 0;
        const int k0 = (lane < 16) ? 0 : 2;
        v2f bm;
        bm.x = (n < 4) ? sP[cam][nn][k0]     : 0.0f;
        bm.y = (n < 4) ? sP[cam][nn][k0 + 1] : 0.0f;

        v8f cz = {};
        v8f d = __builtin_amdgcn_wmma_f32_16x16x4_f32(
            /*neg_a=*/false, a, /*neg_b=*/false, bm,
            /*c_mod=*/(short)0, cz, /*reuse_a=*/false, /*reuse_b=*/false);

        // D 16x16 f32: VGPR r, lane L<16 -> (M=r, N=L); lane 16+L -> (M=8+r, N=L).
        // Column N = component of pc; keep pc0..pc2.
        const int comp = lane & 15;
        if (comp < 3) {
            const int mofs = (lane < 16) ? 0 : 8;
#pragma unroll
            for (int r = 0; r < 8; ++r)
                sPC[cam][mofs + r][comp] = d[r];
        }
    }
    __syncthreads();

    // ---------------- Stage 2: bilinear gather + camera max ----------------
    const float* img_b = images + (size_t)b * NCAM * IMG_H * IMG_W * IMG_C;

#pragma unroll
    for (int pi = wave; pi < TILE; pi += 8) {
        const int gp = base_pix + pi;
        float best = -__builtin_inff();

#pragma unroll
        for (int cam = 0; cam < NCAM; ++cam) {
            const float pc0 = sPC[cam][pi][0];
            const float pc1 = sPC[cam][pi][1];
            const float pc2 = sPC[cam][pi][2];
            const float den = pc2 + 1e-7f;
            const float u = pc0 / den;          // pix_x
            const float v = pc1 / den;          // pix_y

            float x0 = floorf(u), x1 = x0 + 1.0f;
            float y0 = floorf(v), y1 = y0 + 1.0f;
            x0 = fminf(fmaxf(x0, 0.0f), (float)(IMG_W - 1));
            x1 = fminf(fmaxf(x1, 0.0f), (float)(IMG_W - 1));
            y0 = fminf(fmaxf(y0, 0.0f), (float)(IMG_H - 1));
            y1 = fminf(fmaxf(y1, 0.0f), (float)(IMG_H - 1));

            const float wx0 = x1 - u, wx1 = u - x0;   // clip-then-weight (ref)
            const float wy0 = y1 - v, wy1 = v - y0;

            const int i00 = (int)(x0 + y0 * (float)IMG_W);
            const int i01 = (int)(x0 + y1 * (float)IMG_W);
            const int i10 = (int)(x1 + y0 * (float)IMG_W);
            const int i11 = (int)(x1 + y1 * (float)IMG_W);

            const float* base = img_b + (size_t)cam * IMG_H * IMG_W * IMG_C;
            // lane = channel: four 32-lane x 4B = 128B coalesced loads
            const float v00 = base[(size_t)i00 * IMG_C + lane];
            const float v01 = base[(size_t)i01 * IMG_C + lane];
            const float v10 = base[(size_t)i10 * IMG_C + lane];
            const float v11 = base[(size_t)i11 * IMG_C + lane];

            const float r = wx0 * wy0 * v00 + wx0 * wy1 * v01 +
                            wx1 * wy0 * v10 + wx1 * wy1 * v11;
            best = fmaxf(best, r);
        }
        out[((size_t)b * NPIX + gp) * IMG_C + lane] = best;
    }
}

extern "C" void kernel_launch(void* const* d_in, const int* in_sizes, int n_in,
                              void* d_out, int out_size, void* d_ws, size_t ws_size,
                              hipStream_t stream) {
    const float* images = (const float*)d_in[0];
    const float* Ks     = (const float*)d_in[1];
    const float* RTs    = (const float*)d_in[2];
    float* out          = (float*)d_out;

    const int blocks = NBATCH * (NPIX / TILE);   // 2 * 2500 = 5000
    ipm_bev_kernel<<<blocks, 256, 0, stream>>>(images, Ks, RTs, out);
}